// MindPalaceRouter_42984032699160
// MI455X (gfx1250) — compile-verified
//
#include <hip/hip_runtime.h>
#include <math.h>

typedef float v2f __attribute__((ext_vector_type(2)));
typedef float v4f __attribute__((ext_vector_type(4)));
typedef float v8f __attribute__((ext_vector_type(8)));

#define BB   32
#define TT   2048
#define DD   1024
#define NR   64
#define TCH  32          // T chunks for stage-1 reduction

// ---------------- stage 1: partial sums over T (bandwidth-bound, NT loads) ---
// grid: (B=32, TCH=32), block 256 threads. Thread t owns d = 4t..4t+3.
// Deterministic: each block writes its own partial row, no atomics.
__global__ __launch_bounds__(256) void reduce_stage1_kernel(const float* __restrict__ x,
                                                            float* __restrict__ partial) {
    const int b     = blockIdx.x;
    const int chunk = blockIdx.y;
    const int rows  = TT / TCH;                // 64 rows per block
    const int t0    = chunk * rows;
    const int tid   = threadIdx.x;             // 0..255 -> float4 index in row

    const v4f* base = (const v4f*)(x + (size_t)b * TT * DD);
    v4f acc = {0.f, 0.f, 0.f, 0.f};
    for (int r = 0; r < rows; ++r) {
        // x is streamed exactly once (256 MB vs 192 MB L2): non-temporal so the
        // reusable weight matrices stay resident in L2 for the GEMM kernels.
        v4f v = __builtin_nontemporal_load(&base[(size_t)(t0 + r) * (DD / 4) + tid]);
        acc += v;
    }
    v4f* p = (v4f*)(partial + ((size_t)b * TCH + chunk) * DD);
    p[tid] = acc;
}

// ---------------- stage 2: deterministic combine + scale -> mean ----------
// 32768 elements total; thread i -> (b, d), fixed-order sum over 32 chunks.
__global__ __launch_bounds__(256) void reduce_stage2_kernel(const float* __restrict__ partial,
                                                            float* __restrict__ mean) {
    const int i = blockIdx.x * blockDim.x + threadIdx.x;   // 0 .. 32767
    const int b = i >> 10;
    const int d = i & (DD - 1);
    const float* p = partial + (size_t)b * TCH * DD + d;
    float s = 0.0f;
    for (int c = 0; c < TCH; ++c) s += p[(size_t)c * DD];
    mean[i] = s * (1.0f / (float)TT);
}

// ---------------- generic 32 x N x K GEMM via V_WMMA_F32_16X16X4_F32 ----------
// out[m][n] = sum_k A[m][k] * W[n][k] + bias[n]      (A: 32xK, W: NxK row-major)
// One wave: one 16-wide N tile, both 16-row M halves (two v8f accumulators).
// A-frag (16x4 f32): lanes 0-15 -> M=lane, K={k,k+1}; lanes 16-31 -> M=lane-16, K={k+2,k+3}
// B-frag (4x16 f32): lanes 0-15 -> N=lane, K={k,k+1}; lanes 16-31 -> N=lane-16, K={k+2,k+3}
// C/D: VGPR r, lanes 0-15 -> (M=r, N=lane); lanes 16-31 -> (M=r+8, N=lane-16)
// Software-pipelined: next k-step's fragments are loaded before the current
// WMMAs consume theirs, so loads overlap the matrix pipe instead of draining
// loadcnt to zero every step.
__global__ __launch_bounds__(128) void gemm32_wmma_kernel(const float* __restrict__ A,
                                                          const float* __restrict__ W,
                                                          const float* __restrict__ bias,
                                                          float* __restrict__ out,
                                                          int N, int K) {
    const int wave  = threadIdx.x >> 5;
    const int lane  = threadIdx.x & 31;
    const int nTile = blockIdx.x * (blockDim.x >> 5) + wave;
    const int half  = lane >> 4;     // 0: K pair {0,1}, 1: K pair {2,3}
    const int lcol  = lane & 15;     // M row (for A) / N col (for B,C,D)
    const int n     = nTile * 16 + lcol;

    const float* arow0 = A + (size_t)lcol * K + half * 2;         // M rows 0..15
    const float* arow1 = A + (size_t)(lcol + 16) * K + half * 2;  // M rows 16..31
    const float* wrow  = W + (size_t)n * K + half * 2;

    v8f c0 = {};
    v8f c1 = {};

    v2f a0 = *(const v2f*)(arow0);
    v2f a1 = *(const v2f*)(arow1);
    v2f bv = *(const v2f*)(wrow);

#pragma unroll 8
    for (int k = 4; k < K; k += 4) {
        v2f na0 = *(const v2f*)(arow0 + k);
        v2f na1 = *(const v2f*)(arow1 + k);
        v2f nbv = *(const v2f*)(wrow  + k);
        c0 = __builtin_amdgcn_wmma_f32_16x16x4_f32(false, a0, false, bv, (short)0, c0, false, false);
        c1 = __builtin_amdgcn_wmma_f32_16x16x4_f32(false, a1, false, bv, (short)0, c1, false, false);
        a0 = na0; a1 = na1; bv = nbv;
    }
    c0 = __builtin_amdgcn_wmma_f32_16x16x4_f32(false, a0, false, bv, (short)0, c0, false, false);
    c1 = __builtin_amdgcn_wmma_f32_16x16x4_f32(false, a1, false, bv, (short)0, c1, false, false);

    const float bb = bias ? bias[n] : 0.0f;
#pragma unroll
    for (int r = 0; r < 8; ++r) {
        out[(size_t)(r + half * 8)      * N + n] = c0[r] + bb;
        out[(size_t)(16 + r + half * 8) * N + n] = c1[r] + bb;
    }
}

// ---------------- fused softmax / gate / sigmoid ----------------
// grid: 32 blocks (one per batch), 64 threads (one per room m).
__global__ __launch_bounds__(64) void finalize_kernel(const float* __restrict__ ctx,
                                                      const float* __restrict__ warp,   // 32 x 64 x 64 (pre-scale)
                                                      const float* __restrict__ raw,    // 32 x 64
                                                      const float* __restrict__ Wg,     // 64 x 1024
                                                      const float* __restrict__ bg,     // 64
                                                      const float* __restrict__ adjacency, // 64 x 64
                                                      float* __restrict__ out) {
    const int b = blockIdx.x;
    const int m = threadIdx.x;

    // gate linear term: ctx[b] . Wg[m] + bg[m]
    const v4f* c4 = (const v4f*)(ctx + (size_t)b * DD);
    const v4f* w4 = (const v4f*)(Wg  + (size_t)m * DD);
    float gl = bg[m];
    for (int k = 0; k < DD / 4; ++k) {
        v4f cv = c4[k];
        v4f wv = w4[k];
        gl += cv.x * wv.x + cv.y * wv.y + cv.z * wv.z + cv.w * wv.w;
    }

    // softmax over adjacency[m][:] + 0.1*warp[b][m][:], fused with dot(adj_row, raw[b][:])
    const float* wr = warp + ((size_t)b * NR + m) * NR;
    const float* ar = adjacency + (size_t)m * NR;
    const float* rw = raw + (size_t)b * NR;

    float mx = -INFINITY;
    for (int k = 0; k < NR; ++k) {
        float v = ar[k] + 0.1f * wr[k];
        mx = fmaxf(mx, v);
    }
    float sumE = 0.0f, sumR = 0.0f;
    for (int k = 0; k < NR; ++k) {
        float e = expf(ar[k] + 0.1f * wr[k] - mx);
        sumE += e;
        sumR += e * rw[k];
    }
    const float dot = sumR / sumE;

    const float logit = 0.5f * (gl + dot);
    out[(size_t)b * NR + m] = 1.0f / (1.0f + expf(-logit));
}

extern "C" void kernel_launch(void* const* d_in, const int* in_sizes, int n_in,
                              void* d_out, int out_size, void* d_ws, size_t ws_size,
                              hipStream_t stream) {
    const float* x         = (const float*)d_in[0];   // 32 x 2048 x 1024
    const float* summaries = (const float*)d_in[1];   // 64 x 1024
    const float* Wc        = (const float*)d_in[2];   // 1024 x 1024
    const float* bc        = (const float*)d_in[3];   // 1024
    const float* Wg        = (const float*)d_in[4];   // 64 x 1024
    const float* bg        = (const float*)d_in[5];   // 64
    const float* Ww        = (const float*)d_in[6];   // 4096 x 1024
    const float* bw        = (const float*)d_in[7];   // 4096
    const float* adjacency = (const float*)d_in[8];   // 64 x 64
    float* out = (float*)d_out;                       // 32 x 64

    float* ws      = (float*)d_ws;
    float* partial = ws;                          // 32*32*1024 = 1,048,576 floats (4 MB)
    float* mean    = partial + BB * TCH * DD;     // 32*1024
    float* ctx     = mean + BB * DD;              // 32*1024
    float* warp    = ctx  + BB * DD;              // 32*4096
    float* raw     = warp + BB * NR * NR;         // 32*64

    // 1) partial sums over T (dominant: 256 MB NT stream -> ~11us @ 23.3 TB/s)
    reduce_stage1_kernel<<<dim3(BB, TCH), 256, 0, stream>>>(x, partial);

    // 2) deterministic combine -> mean
    reduce_stage2_kernel<<<(BB * DD + 255) / 256, 256, 0, stream>>>(partial, mean);

    // 3) ctx = mean @ Wc^T + bc        (N=1024, 64 tiles, 4 waves/block -> 16 blocks)
    gemm32_wmma_kernel<<<16, 128, 0, stream>>>(mean, Wc, bc, ctx, DD, DD);

    // 4) warp = ctx @ Ww^T + bw        (N=4096, 256 tiles -> 64 blocks)
    gemm32_wmma_kernel<<<64, 128, 0, stream>>>(ctx, Ww, bw, warp, BB * 128, DD);

    // 5) raw = ctx @ summaries^T       (N=64, 4 tiles -> 1 block)
    gemm32_wmma_kernel<<<1, 128, 0, stream>>>(ctx, summaries, nullptr, raw, NR, DD);

    // 6) softmax + gate + sigmoid
    finalize_kernel<<<BB, NR, 0, stream>>>(ctx, warp, raw, Wg, bg, adjacency, out);
}